// KernelConv_2886218022961
// MI455X (gfx1250) — compile-verified
//
#include <hip/hip_runtime.h>
#include <stdint.h>

typedef float v4f __attribute__((ext_vector_type(4)));

#define KS 5
#define BB 4
#define CC 3
#define HH 512
#define WW 512
#define ROWLEN 520   // floats per LDS row (pads + 512 data + alignment slack)
#define DOFF 4       // frame column w sits at LDS index DOFF + w (keeps b128 stores 16B-aligned)

// One block = one output row (b, c, h). 128 threads x 4 pixels = 512 pixels.
// Frames rows h-2..h+2 staged into LDS with CDNA5 async global->LDS copies;
// core streamed once with non-temporal b128 loads (protects L2-resident frames).
__global__ __launch_bounds__(128) void kpn_adaptive_conv_kernel(
    const float* __restrict__ frames,
    const float* __restrict__ core,
    float* __restrict__ out)
{
    __shared__ __align__(16) float smem[KS * ROWLEN];

    const int tid = threadIdx.x;           // 0..127
    const int bid = blockIdx.x;            // (b*CC + c)*HH + h
    const int h   = bid & (HH - 1);
    const int bc  = bid >> 9;              // b*CC + c
    const int c   = bc % CC;
    const int b   = bc / CC;
    const int w0  = tid << 2;              // first of 4 pixels owned by this lane

    // ---------------- Stage frame rows into LDS (async tensor/LDS path) ----------------
    const uint32_t lds_base = (uint32_t)(uintptr_t)&smem[0];
    const float* frame_plane = frames + (size_t)bc * (HH * WW);

#pragma unroll
    for (int i = 0; i < KS; ++i) {
        const int g = h + i - 2;           // uniform across the block -> uniform branch
        if ((unsigned)g < (unsigned)HH) {
            const uint32_t lds_addr = lds_base + (uint32_t)((i * ROWLEN + DOFF + w0) * 4);
            const uint64_t gaddr    = (uint64_t)(uintptr_t)(frame_plane + (size_t)g * WW + w0);
            // CDNA5: GLOBAL_LOAD_ASYNC_TO_LDS_B128 — copies 16 bytes from global memory into LDS, tracked by ASYNCcnt
            asm volatile("global_load_async_to_lds_b128 %0, %1, off"
                         :: "v"(lds_addr), "v"(gaddr) : "memory");
        } else {
            // zero-padded row (jnp.pad semantics)
            *(v4f*)&smem[i * ROWLEN + DOFF + w0] = (v4f)0.0f;
        }
    }
    // halo pads: w = -2,-1 and w = 512,513 for every row
    if (tid < KS) {
        float* r = &smem[tid * ROWLEN];
        r[DOFF - 2]      = 0.0f;
        r[DOFF - 1]      = 0.0f;
        r[DOFF + WW]     = 0.0f;
        r[DOFF + WW + 1] = 0.0f;
    }

    asm volatile("s_wait_asynccnt 0" ::: "memory");
    __syncthreads();

    // ---------------- Cache the 12-float LDS window per row into registers ----------------
    // Needed frame value for (tap j, subpixel k): column w0+k+j-2  -> LDS index i*ROWLEN + w0 + (k+j+2)
    float fr[KS][12];
#pragma unroll
    for (int i = 0; i < KS; ++i) {
        const v4f a0 = *(const v4f*)&smem[i * ROWLEN + w0];
        const v4f a1 = *(const v4f*)&smem[i * ROWLEN + w0 + 4];
        const v4f a2 = *(const v4f*)&smem[i * ROWLEN + w0 + 8];
        fr[i][0] = a0.x; fr[i][1] = a0.y; fr[i][2]  = a0.z; fr[i][3]  = a0.w;
        fr[i][4] = a1.x; fr[i][5] = a1.y; fr[i][6]  = a1.z; fr[i][7]  = a1.w;
        fr[i][8] = a2.x; fr[i][9] = a2.y; fr[i][10] = a2.z; fr[i][11] = a2.w;
    }

    // ---------------- Stream core once (NT loads), 25 FMAs per pixel ----------------
    // core index: (((b*25 + t)*3 + c)*512 + h)*512 + w  = base + t * (3*512*512)
    const size_t core_base = (((size_t)(b * 25) * CC + c) * HH + h) * WW + (size_t)w0;
    const size_t tap_stride = (size_t)CC * HH * WW;

    float ax = 0.0f, ay = 0.0f, az = 0.0f, aw = 0.0f;
#pragma unroll
    for (int t = 0; t < KS * KS; ++t) {
        const int i = t / KS;
        const int j = t % KS;
        const v4f cv = __builtin_nontemporal_load(
            (const v4f*)(core + core_base + (size_t)t * tap_stride));
        ax += cv.x * fr[i][j + 2];
        ay += cv.y * fr[i][j + 3];
        az += cv.z * fr[i][j + 4];
        aw += cv.w * fr[i][j + 5];
    }

    v4f res; res.x = ax; res.y = ay; res.z = az; res.w = aw;
    float* optr = out + ((size_t)bc * HH + h) * WW + (size_t)w0;
    __builtin_nontemporal_store(res, (v4f*)optr);
}

extern "C" void kernel_launch(void* const* d_in, const int* in_sizes, int n_in,
                              void* d_out, int out_size, void* d_ws, size_t ws_size,
                              hipStream_t stream) {
    const float* frames = (const float*)d_in[0];   // [4,1,3,512,512] fp32
    const float* core   = (const float*)d_in[1];   // [4,1,25,3,512,512] fp32
    float* out          = (float*)d_out;           // [4,3,512,512] fp32

    const int grid = BB * CC * HH;                 // 6144 blocks, one per output row
    kpn_adaptive_conv_kernel<<<grid, 128, 0, stream>>>(frames, core, out);
}